// CIDL_RNN_11527692222526
// MI455X (gfx1250) — compile-verified
//
#include <hip/hip_runtime.h>
#include <cmath>

// ---------------------------------------------------------------------------
// Persistent fused SRC-LSTM for MI455X (gfx1250, wave32, WMMA f32 16x16x4).
//
// Shapes (hardcoded from the reference):
//   x        : (B=32, T=8192, D=4)   fp32
//   weight_ih: (4H=256, D=4)         fp32
//   weight_hh: (4H=256, H=64)        fp32
//   bias_ih  : (256,), bias_hh: (256,)
//   out      : [h | h | c], each (32, 8192, 64) fp32, concatenated flat.
//
// Grid: 2 blocks (16 batch rows each, one per WGP). Block: 16 waves.
// Wave w owns gate columns [16w, 16w+16):
//   acc = WMMA(x_frag, Wih_frag, bias)                (K=4 == D, one op)
//   16x acc = WMMA(hread_frag, Whh_frag[ks], acc)     (K=64 recurrent GEMM)
// Then gates -> LDS, barrier, elementwise LSTM update + 4-tap interpolation
// of the next h_read fused into one pass, barrier, next step.
//
// All activations are branch-free v_exp_f32 + v_rcp_f32 sequences (no IEEE
// division on the serial critical path).
// ---------------------------------------------------------------------------

typedef __attribute__((ext_vector_type(2))) float v2f;
typedef __attribute__((ext_vector_type(8))) float v8f;

#define B_TOT   32
#define T_LEN   8192
#define D_IN    4
#define HID     64
#define NG      256          // 4*HID
#define MTILE   16
#define NWAVES  16
#define THREADS 512
#define BTH     ((size_t)B_TOT * T_LEN * HID)   // 16,777,216 elements per output

// interp kernel for OS_FACTOR = 48000/44100 (< 2 branch: delta = os-1, eps=1)
static constexpr double OSF   = 48000.0 / 44100.0;
static constexpr double DLT   = OSF - 1.0;
static constexpr float  KRN0  = (float)(-(DLT - 1.0) * (DLT - 2.0) * (DLT - 3.0) / 6.0);
static constexpr float  KRN1  = (float)( DLT * (DLT - 2.0) * (DLT - 3.0) / 2.0);
static constexpr float  KRN2  = (float)(-DLT * (DLT - 1.0) * (DLT - 3.0) / 2.0);
static constexpr float  KRN3  = (float)( DLT * (DLT - 1.0) * (DLT - 2.0) / 6.0);

__device__ __forceinline__ float fast_rcp(float x) {
    return __builtin_amdgcn_rcpf(x);              // v_rcp_f32 (no NR refinement)
}
__device__ __forceinline__ float fast_tanh(float x) {
    // 1 - 2/(e^{2x}+1); saturates correctly at +/-inf of expf
    return 1.0f - 2.0f * fast_rcp(__expf(2.0f * x) + 1.0f);
}

__global__ __launch_bounds__(THREADS)
void srclstm_persistent_kernel(const float* __restrict__ x,
                               const float* __restrict__ w_ih,
                               const float* __restrict__ w_hh,
                               const float* __restrict__ b_ih,
                               const float* __restrict__ b_hh,
                               float* __restrict__ out) {
    // Padded LDS to avoid bank conflicts:
    //  gates row stride 260  -> rows r and r+8 land on different bank sets
    //  hread row stride 132  -> 16 A-lanes hit 16 distinct banks
    __shared__ float gates[MTILE][NG + 4];        // ~16.6 KB
    __shared__ float hist [4][MTILE][2 * HID];    //  32   KB state ring (h|c)
    __shared__ float hread[MTILE][2 * HID + 4];   //  ~8.4 KB interpolated state

    const int tid  = threadIdx.x;
    const int wave = tid >> 5;
    const int lane = tid & 31;
    const int b0   = blockIdx.x * MTILE;

    const int tileN = wave * 16;
    const int col   = tileN + (lane & 15);        // this lane's gate column (B/C/D N)
    const int m_a   = lane & 15;                  // this lane's A-matrix row
    const int koff  = (lane < 16) ? 0 : 2;        // A/B K offset per 16x16x4 layout

    // --- persistent register-resident weight fragments --------------------
    v2f bhh[16];                                  // W_hh B-fragments, 16 K-steps
#pragma unroll
    for (int ks = 0; ks < 16; ++ks) {
        const float* p = w_hh + (size_t)col * HID + 4 * ks + koff;
        bhh[ks].x = p[0];
        bhh[ks].y = p[1];
    }
    v2f bih;                                      // W_ih B-fragment (K=4 total)
    {
        const float* p = w_ih + (size_t)col * D_IN + koff;
        bih.x = p[0];
        bih.y = p[1];
    }
    const float bias = b_ih[col] + b_hh[col];
    v8f cinit;
#pragma unroll
    for (int i = 0; i < 8; ++i) cinit[i] = bias;  // C/D layout: every VGPR holds col N

    // Branch-free activation parameters:
    //   y = rcp(1 + exp(-pm*x));  result = fma(pm, y, 1-pm)
    //   pm = 1 -> sigmoid(x);  pm = 2 -> 2*sigmoid(2x)-1 = tanh(x)
    const bool  is_g_gate = ((wave >> 2) == 2);   // waves 8..11 -> tanh gate
    const float pm = is_g_gate ? 2.0f : 1.0f;
    const float pc = 1.0f - pm;

    // --- zero-init LDS state ----------------------------------------------
    {
        float* h0 = &hread[0][0];
        for (int i = tid; i < MTILE * (2 * HID + 4); i += THREADS) h0[i] = 0.0f;
        float* h1 = &hist[0][0][0];
        for (int i = tid; i < 4 * MTILE * 2 * HID; i += THREADS) h1[i] = 0.0f;
    }
    __syncthreads();

    const float* xp = x + (size_t)(b0 + m_a) * (T_LEN * D_IN) + koff;

    for (int t = 0; t < T_LEN; ++t) {
        // ---------------- GEMM phase (per-wave 16x16 gate tile) -----------
        v2f xa;
        xa.x = xp[0];
        xa.y = xp[1];
        __builtin_prefetch(xp + D_IN, 0, 3);      // speculative global_prefetch_b8
        xp += D_IN;

        // gates_tile = x_t @ W_ih^T + bias   (single K=4 WMMA)
        v8f acc = __builtin_amdgcn_wmma_f32_16x16x4_f32(
            false, xa, false, bih, (short)0, cinit, false, false);

        // gates_tile += h_interp @ W_hh^T    (K=64 as 16 x K=4 WMMAs)
#pragma unroll
        for (int ks = 0; ks < 16; ++ks) {
            v2f a;
            a.x = hread[m_a][4 * ks + koff];
            a.y = hread[m_a][4 * ks + koff + 1];
            acc = __builtin_amdgcn_wmma_f32_16x16x4_f32(
                false, a, false, bhh[ks], (short)0, acc, false, false);
        }

        // activations + spill tile to LDS (C/D layout: VGPR r -> row r / r+8)
        const int rbase = (lane < 16) ? 0 : 8;
#pragma unroll
        for (int r = 0; r < 8; ++r) {
            const float v = acc[r];
            const float y = fast_rcp(1.0f + __expf(-pm * v));
            gates[r + rbase][col] = fmaf(pm, y, pc);
        }
        __syncthreads();

        // ---------------- elementwise LSTM update + interpolation ---------
        const int s0 = t & 3;                 // slot for state[t] (evicts t-4)
        const int s1 = (t + 3) & 3;           // state[t-1]
        const int s2 = (t + 2) & 3;           // state[t-2]
        const int s3 = (t + 1) & 3;           // state[t-3]
#pragma unroll
        for (int e = 0; e < 2; ++e) {
            const int idx = tid + e * THREADS;      // 1024 (m,n) elements total
            const int m = idx >> 6;
            const int n = idx & 63;

            const float ig = gates[m][n];
            const float fg = gates[m][n + 64];
            const float gg = gates[m][n + 128];
            const float og = gates[m][n + 192];
            const float ci = hread[m][HID + n];     // interpolated c (own element)

            const float c_new = fg * ci + ig * gg;
            const float h_new = og * fast_tanh(c_new);

            hist[s0][m][n]       = h_new;
            hist[s0][m][HID + n] = c_new;

            // h_read for step t+1: k0*state[t] + k1..k3 * older ring slots
            const float hr = KRN0 * h_new
                           + KRN1 * hist[s1][m][n]
                           + KRN2 * hist[s2][m][n]
                           + KRN3 * hist[s3][m][n];
            const float cr = KRN0 * c_new
                           + KRN1 * hist[s1][m][HID + n]
                           + KRN2 * hist[s2][m][HID + n]
                           + KRN3 * hist[s3][m][HID + n];
            hread[m][n]       = hr;                 // safe: GEMM reads done (barrier)
            hread[m][HID + n] = cr;                 // safe: only this thread touches it

            const size_t o = ((size_t)(b0 + m) * T_LEN + t) * HID + n;
            out[o]           = h_new;               // output 0: h
            out[BTH + o]     = h_new;               // output 1: h (duplicated)
            out[2 * BTH + o] = c_new;               // output 2: c
        }
        __syncthreads();
    }
}

extern "C" void kernel_launch(void* const* d_in, const int* in_sizes, int n_in,
                              void* d_out, int out_size, void* d_ws, size_t ws_size,
                              hipStream_t stream) {
    const float* x    = (const float*)d_in[0];
    const float* w_ih = (const float*)d_in[1];
    const float* w_hh = (const float*)d_in[2];
    const float* b_ih = (const float*)d_in[3];
    const float* b_hh = (const float*)d_in[4];
    float* out = (float*)d_out;

    dim3 grid(B_TOT / MTILE);   // 2 persistent blocks, one WGP each
    srclstm_persistent_kernel<<<grid, THREADS, 0, stream>>>(
        x, w_ih, w_hh, b_ih, b_hh, out);
}